// YOLOv2_16544214024247
// MI455X (gfx1250) — compile-verified
//
#include <hip/hip_runtime.h>
#include <hip/hip_bf16.h>

// YOLOv2 post-process for MI455X (gfx1250, wave32).
// Pipeline: K0 zero-init -> K1 score histogram (streaming pass 1, coalesced b128)
//        -> K2 threshold select -> K3 candidate compaction (streaming pass 2)
//        -> K4 top-100 extraction (wave32 shfl) + box decode
//           + V_WMMA_F32_16X16X4_F32 pairwise area-sums + divide-free IoU NMS.

#define NUM_CLASSES   20
#define STRIDE_PX     32.0f
#define TOPK          100
#define TOPK_PAD      112          // 7 tiles of 16 for WMMA
#define CONF_THRESH   0.01f
#define NMS_THRESH    0.5f
#define CLASS_OFFSET  1000000.0f
#define HIST_BINS     2048         // float bits >> 19 (exp + 4 mantissa bits), scores in (0,1)
#define CAND_MAX      4096

typedef float v2f __attribute__((ext_vector_type(2)));
typedef float v8f __attribute__((ext_vector_type(8)));

struct Ws {
    unsigned hist[HIST_BINS];
    unsigned threshKey;
    unsigned candCount;
    float    candVal[CAND_MAX];
    unsigned candIdx[CAND_MAX];
};

__device__ __forceinline__ float sigf(float x) { return 1.0f / (1.0f + __expf(-x)); }

// ---------------- K0: zero workspace (re-run every launch: graph-replay safe) ----
__global__ void k0_init(Ws* __restrict__ ws) {
    int i = blockIdx.x * blockDim.x + threadIdx.x;
    if (i < HIST_BINS) ws->hist[i] = 0u;
    if (i == 0) { ws->candCount = 0u; ws->threshKey = 0u; }
}

// ---------------- K1: histogram of score = sigmoid(obj)*sigmoid(cls) ------------
// One float4 of cls per thread-iteration: consecutive lanes -> consecutive
// global_load_b128 (512B/wave, perfectly coalesced). 20 % 4 == 0, so every
// float4 stays inside one anchor's class row; obj[i4/5] is L0/L2-resident.
__global__ __launch_bounds__(256) void k1_hist(const float* __restrict__ obj,
                                               const float* __restrict__ cls,
                                               Ws* __restrict__ ws, int M) {
    __shared__ unsigned h[HIST_BINS];
    for (int i = threadIdx.x; i < HIST_BINS; i += blockDim.x) h[i] = 0u;
    __syncthreads();
    const float4* __restrict__ cls4 = (const float4*)cls;
    const int n4 = M * 5;                        // total float4s in cls
    const int stride = gridDim.x * blockDim.x;
    for (int i4 = blockIdx.x * blockDim.x + threadIdx.x; i4 < n4; i4 += stride) {
        __builtin_prefetch(cls4 + i4 + stride, 0, 0);   // -> global_prefetch_b8
        float so = sigf(obj[i4 / 5]);
        float4 c = cls4[i4];
        float s0 = so * sigf(c.x);
        float s1 = so * sigf(c.y);
        float s2 = so * sigf(c.z);
        float s3 = so * sigf(c.w);
        atomicAdd(&h[min(__float_as_uint(s0) >> 19, (unsigned)(HIST_BINS - 1))], 1u);
        atomicAdd(&h[min(__float_as_uint(s1) >> 19, (unsigned)(HIST_BINS - 1))], 1u);
        atomicAdd(&h[min(__float_as_uint(s2) >> 19, (unsigned)(HIST_BINS - 1))], 1u);
        atomicAdd(&h[min(__float_as_uint(s3) >> 19, (unsigned)(HIST_BINS - 1))], 1u);
    }
    __syncthreads();
    for (int i = threadIdx.x; i < HIST_BINS; i += blockDim.x)
        if (h[i]) atomicAdd(&ws->hist[i], h[i]);
}

// ---------------- K2: pick largest bin b with suffix-count >= TOPK --------------
__global__ void k2_thresh(Ws* __restrict__ ws) {
    if (threadIdx.x == 0 && blockIdx.x == 0) {
        unsigned cum = 0u, key = 0u;
        for (int b = HIST_BINS - 1; b >= 0; --b) {
            cum += ws->hist[b];
            if (cum >= (unsigned)TOPK) { key = ((unsigned)b) << 19; break; }
        }
        ws->threshKey = key;   // float bits >= key  <=>  score in/above threshold bin
    }
}

// ---------------- K3: compact candidates above threshold -------------------------
__global__ __launch_bounds__(256) void k3_compact(const float* __restrict__ obj,
                                                  const float* __restrict__ cls,
                                                  Ws* __restrict__ ws, int M) {
    const unsigned key = ws->threshKey;
    const float4* __restrict__ cls4 = (const float4*)cls;
    const int n4 = M * 5;
    const int stride = gridDim.x * blockDim.x;
    for (int i4 = blockIdx.x * blockDim.x + threadIdx.x; i4 < n4; i4 += stride) {
        __builtin_prefetch(cls4 + i4 + stride, 0, 0);
        float so = sigf(obj[i4 / 5]);
        float4 c = cls4[i4];
        float s[4] = { so * sigf(c.x), so * sigf(c.y), so * sigf(c.z), so * sigf(c.w) };
#pragma unroll
        for (int e = 0; e < 4; ++e) {
            if (__float_as_uint(s[e]) >= key) {
                unsigned p = atomicAdd(&ws->candCount, 1u);
                if (p < CAND_MAX) {
                    ws->candVal[p] = s[e];
                    ws->candIdx[p] = (unsigned)i4 * 4u + (unsigned)e;  // flat score index
                }
            }
        }
    }
}

// ---------------- K4: top-100, decode, WMMA area-sums, IoU + NMS -----------------
__global__ __launch_bounds__(1024) void k4_final(const float* __restrict__ reg,
                                                 const float* __restrict__ anc,
                                                 Ws* __restrict__ ws,
                                                 float* __restrict__ out) {
    __shared__ float    lval[CAND_MAX];
    __shared__ unsigned lidx[CAND_MAX];
    __shared__ float    wV[32];
    __shared__ unsigned wI[32];
    __shared__ int      wS[32];
    __shared__ float    topV[TOPK_PAD];
    __shared__ unsigned topI[TOPK_PAD];
    __shared__ float    X1[TOPK_PAD], Y1[TOPK_PAD], X2[TOPK_PAD], Y2[TOPK_PAD], AREA[TOPK_PAD];
    __shared__ unsigned sup[TOPK_PAD * 4];   // 128-bit suppression row per box
    __shared__ unsigned keepMask[4];

    const int tid  = threadIdx.x;
    const int wave = tid >> 5;
    const int lane = tid & 31;               // wave32 on gfx1250
    const int n = (int)min(ws->candCount, (unsigned)CAND_MAX);

    for (int i = tid; i < CAND_MAX; i += 1024) {
        lval[i] = (i < n) ? ws->candVal[i] : -1.0f;
        lidx[i] = (i < n) ? ws->candIdx[i] : 0xFFFFFFFFu;
    }
    for (int i = tid; i < TOPK_PAD * 4; i += 1024) sup[i] = 0u;
    __syncthreads();

    // Iterative max extraction: (value desc, index asc) matches jax.lax.top_k.
    // Wave32 shuffle reduction (5 steps) + one 32-entry cross-wave step.
    for (int k = 0; k < TOPK; ++k) {
        float bv = -2.0f; unsigned bi = 0xFFFFFFFFu; int bs = -1;
        for (int i = tid; i < CAND_MAX; i += 1024) {
            float v = lval[i]; unsigned id = lidx[i];
            if (v > bv || (v == bv && id < bi)) { bv = v; bi = id; bs = i; }
        }
#pragma unroll
        for (int off = 16; off > 0; off >>= 1) {
            float    v2 = __shfl_down(bv, off, 32);
            unsigned i2 = __shfl_down(bi, off, 32);
            int      s2 = __shfl_down(bs, off, 32);
            if (v2 > bv || (v2 == bv && i2 < bi)) { bv = v2; bi = i2; bs = s2; }
        }
        if (lane == 0) { wV[wave] = bv; wI[wave] = bi; wS[wave] = bs; }
        __syncthreads();
        if (wave == 0) {
            bv = wV[lane]; bi = wI[lane]; bs = wS[lane];
#pragma unroll
            for (int off = 16; off > 0; off >>= 1) {
                float    v2 = __shfl_down(bv, off, 32);
                unsigned i2 = __shfl_down(bi, off, 32);
                int      s2 = __shfl_down(bs, off, 32);
                if (v2 > bv || (v2 == bv && i2 < bi)) { bv = v2; bi = i2; bs = s2; }
            }
            if (lane == 0) {
                topV[k] = bv;
                topI[k] = (bi == 0xFFFFFFFFu) ? 0u : bi;
                if (bs >= 0) lval[bs] = -2.0f;   // remove winner
            }
        }
        __syncthreads();
    }

    // Decode top-100 boxes; stage class-offset coords + areas for NMS.
    if (tid < TOPK_PAD) {
        int k = tid;
        if (k < TOPK) {
            unsigned idx = topI[k];
            unsigned a   = idx / NUM_CLASSES;
            unsigned lab = idx % NUM_CLASSES;
            float4 r  = ((const float4*)reg)[a];
            float4 an = ((const float4*)anc)[a];
            float cx = (sigf(r.x) + an.x) * STRIDE_PX;
            float cy = (sigf(r.y) + an.y) * STRIDE_PX;
            float w  = __expf(r.z) * an.z;
            float h  = __expf(r.w) * an.w;
            float x1 = cx - 0.5f * w, y1 = cy - 0.5f * h;
            float x2 = cx + 0.5f * w, y2 = cy + 0.5f * h;
            out[k * 4 + 0] = x1; out[k * 4 + 1] = y1;
            out[k * 4 + 2] = x2; out[k * 4 + 3] = y2;
            out[400 + k] = topV[k];
            out[500 + k] = (float)lab;
            float off = (float)lab * CLASS_OFFSET;   // per-class NMS via coordinate offset
            float bx1 = x1 + off, by1 = y1 + off, bx2 = x2 + off, by2 = y2 + off;
            X1[k] = bx1; Y1[k] = by1; X2[k] = bx2; Y2[k] = by2;
            AREA[k] = (bx2 - bx1) * (by2 - by1);
        } else {
            X1[k] = 0.0f; Y1[k] = 0.0f; X2[k] = 0.0f; Y2[k] = 0.0f; AREA[k] = 0.0f;
        }
    }
    __syncthreads();

    // Pairwise area sums areas[i]+areas[j] as a rank-2 matmul:
    //   A(16x4) row m = [area_m, 1, 0, 0],  B(4x16) col n = [1, area_n, 0, 0]^T
    // via V_WMMA_F32_16X16X4_F32 (wave32; EXEC all-ones at the call).
    for (int t = wave; t < 49; t += 32) { // 7x7 tiles over 112 padded boxes, 32 waves
        int ti = t / 7, tj = t % 7;
        v2f Aa, Bb;
        if (lane < 16) {
            Aa.x = AREA[ti * 16 + lane]; Aa.y = 1.0f;        // K=0, K=1
            Bb.x = 1.0f;                 Bb.y = AREA[tj * 16 + lane];
        } else {
            Aa.x = 0.0f; Aa.y = 0.0f;                        // K=2, K=3 unused
            Bb.x = 0.0f; Bb.y = 0.0f;
        }
        v8f D = {};
        D = __builtin_amdgcn_wmma_f32_16x16x4_f32(false, Aa, false, Bb,
                                                  (short)0, D, false, false);
        int j       = tj * 16 + (lane & 15);
        int rowBase = ti * 16 + ((lane >= 16) ? 8 : 0);
#pragma unroll
        for (int r = 0; r < 8; ++r) {
            int i = rowBase + r;
            float asum = D[r];                       // area_i + area_j
            float xx1 = fmaxf(X1[i], X1[j]);
            float yy1 = fmaxf(Y1[i], Y1[j]);
            float xx2 = fminf(X2[i], X2[j]);
            float yy2 = fminf(Y2[i], Y2[j]);
            float w = fmaxf(1e-10f, xx2 - xx1);
            float h = fmaxf(1e-10f, yy2 - yy1);
            float inter = w * h;
            float denom = asum - inter;
            // Divide-free:  inter/denom > T  <=>
            //   denom > 0 : inter > T*denom        (cross-multiply)
            //   denom == 0: +inf > T  -> suppress  (inter > 0 always holds)
            //   denom < 0 : negative iou -> never suppress
            if (denom >= 0.0f && inter > NMS_THRESH * denom &&
                i < TOPK && j < TOPK && i != j)
                atomicOr(&sup[i * 4 + (j >> 5)], 1u << (j & 31));
        }
    }
    __syncthreads();

    // Sequential greedy NMS on 128-bit masks (thread 0; trivially cheap).
    if (tid == 0) {
        unsigned keep[4];
        for (int w = 0; w < 4; ++w) {
            unsigned m = 0u;
            for (int b = 0; b < 32; ++b) {
                int k = w * 32 + b;
                if (k < TOPK && topV[k] > CONF_THRESH) m |= (1u << b);
            }
            keep[w] = m;
        }
        for (int i = 0; i < TOPK; ++i) {
            if ((keep[i >> 5] >> (i & 31)) & 1u) {
                for (int w = 0; w < 4; ++w) {
                    unsigned s = sup[i * 4 + w];
                    if (w < (i >> 5))       s = 0u;                         // j < i words
                    else if (w == (i >> 5)) s &= ~((2u << (i & 31)) - 1u);  // clear j <= i
                    keep[w] &= ~s;
                }
            }
        }
        keepMask[0] = keep[0]; keepMask[1] = keep[1];
        keepMask[2] = keep[2]; keepMask[3] = keep[3];
    }
    __syncthreads();
    if (tid < TOPK)
        out[600 + tid] = ((keepMask[tid >> 5] >> (tid & 31)) & 1u) ? 1.0f : 0.0f;
}

// ---------------- host launcher --------------------------------------------------
extern "C" void kernel_launch(void* const* d_in, const int* in_sizes, int n_in,
                              void* d_out, int out_size, void* d_ws, size_t ws_size,
                              hipStream_t stream) {
    const float* obj = (const float*)d_in[0];   // (M,1)
    const float* cls = (const float*)d_in[1];   // (M,20)
    const float* reg = (const float*)d_in[2];   // (M,4)
    const float* anc = (const float*)d_in[3];   // (M,4)
    float* out = (float*)d_out;                 // 400 bbox + 100 score + 100 label + 100 keep
    Ws* ws = (Ws*)d_ws;
    const int M = in_sizes[0];

    k0_init   <<<(HIST_BINS + 255) / 256, 256, 0, stream>>>(ws);
    k1_hist   <<<2048, 256, 0, stream>>>(obj, cls, ws, M);
    k2_thresh <<<1, 1, 0, stream>>>(ws);
    k3_compact<<<2048, 256, 0, stream>>>(obj, cls, ws, M);
    k4_final  <<<1, 1024, 0, stream>>>(reg, anc, ws, out);
}